// sub_channel_attention_80178449482352
// MI455X (gfx1250) — compile-verified
//
#include <hip/hip_runtime.h>
#include <math.h>

typedef __attribute__((ext_vector_type(2))) float v2f;
typedef __attribute__((ext_vector_type(8))) float v8f;

#define BATCH 8
#define C_IN  1024
#define C2    512
#define RBOT  73
#define H28   28
#define HW28  784
#define H56   56
#define HW56  3136
#define BN_EPS 1e-5f

// ---------------------------------------------------------------------------
// V_WMMA_F32_16X16X4_F32 wrapper (D = A(16x4) * B(4x16) + C, f32 throughout)
// ---------------------------------------------------------------------------
__device__ __forceinline__ v8f wmma4(v2f a, v2f b, v8f c) {
  return __builtin_amdgcn_wmma_f32_16x16x4_f32(
      /*neg_a=*/false, a, /*neg_b=*/false, b,
      /*c_mod=*/(short)0, c, /*reuse_a=*/false, /*reuse_b=*/false);
}

// Async global->LDS copies (CDNA5, ASYNCcnt-tracked).
__device__ __forceinline__ void async_copy_b128(uint32_t lds_off, const float* gptr) {
  asm volatile("global_load_async_to_lds_b128 %0, %1, off"
               :: "v"(lds_off), "v"(gptr) : "memory");
}
__device__ __forceinline__ void async_copy_b32(uint32_t lds_off, const float* gptr) {
  asm volatile("global_load_async_to_lds_b32 %0, %1, off"
               :: "v"(lds_off), "v"(gptr) : "memory");
}
__device__ __forceinline__ uint32_t lds_off32(const void* p) {
  // generic LDS pointer: addr[31:0] is the wave-relative LDS byte offset
  return (uint32_t)(uintptr_t)p;
}

// ---------------------------------------------------------------------------
// 1) Global average pool over y: (b,C2,56,56) -> s_gap (b,C2)
// ---------------------------------------------------------------------------
__global__ void gap_kernel(const float* __restrict__ y, float* __restrict__ s_gap) {
  __shared__ float red[256];
  int bc = blockIdx.x; // b*C2 + c
  const float* p = y + (size_t)bc * HW56;
  float sum = 0.f;
  for (int i = threadIdx.x; i < HW56; i += 256) sum += p[i];
  red[threadIdx.x] = sum;
  __syncthreads();
  for (int s = 128; s > 0; s >>= 1) {
    if (threadIdx.x < s) red[threadIdx.x] += red[threadIdx.x + s];
    __syncthreads();
  }
  if (threadIdx.x == 0) s_gap[bc] = red[0] * (1.f / (float)HW56);
}

// ---------------------------------------------------------------------------
// 2) SE FCs: s_scale = sigmoid( fc2 @ relu(fc1 @ s_gap) ), per batch
// ---------------------------------------------------------------------------
__global__ void se_fc_kernel(const float* __restrict__ s_gap,
                             const float* __restrict__ fc1_w,  // (73,512)
                             const float* __restrict__ fc2_w,  // (512,73)
                             float* __restrict__ s_scale) {
  __shared__ float sv[C2];
  __shared__ float hv[RBOT];
  int b = blockIdx.x;
  for (int i = threadIdx.x; i < C2; i += 256) sv[i] = s_gap[b * C2 + i];
  __syncthreads();
  if (threadIdx.x < RBOT) {
    const float* wr = fc1_w + threadIdx.x * C2;
    float acc = 0.f;
    for (int c = 0; c < C2; ++c) acc += wr[c] * sv[c];
    hv[threadIdx.x] = acc > 0.f ? acc : 0.f;
  }
  __syncthreads();
  for (int c = threadIdx.x; c < C2; c += 256) {
    const float* wr = fc2_w + c * RBOT;
    float acc = 0.f;
    for (int r = 0; r < RBOT; ++r) acc += wr[r] * hv[r];
    s_scale[b * C2 + c] = 1.f / (1.f + expf(-acc));
  }
}

// ---------------------------------------------------------------------------
// 3) Depthwise 3x3 conv on x: (b,C,28,28) -> x1
// ---------------------------------------------------------------------------
__global__ void dwconv_kernel(const float* __restrict__ x,
                              const float* __restrict__ dw_w,  // (C,1,3,3)
                              const float* __restrict__ dw_b,
                              float* __restrict__ x1) {
  int bc = blockIdx.x; // b*C + c
  int c = bc % C_IN;
  const float* xp = x + (size_t)bc * HW28;
  float* op = x1 + (size_t)bc * HW28;
  float w[9];
#pragma unroll
  for (int j = 0; j < 9; ++j) w[j] = dw_w[c * 9 + j];
  float bias = dw_b[c];
  for (int p = threadIdx.x; p < HW28; p += 256) {
    int h = p / H28, wq = p % H28;
    float acc = bias;
#pragma unroll
    for (int kh = 0; kh < 3; ++kh) {
      int hh = h + kh - 1;
      if (hh < 0 || hh >= H28) continue;
#pragma unroll
      for (int kw = 0; kw < 3; ++kw) {
        int ww = wq + kw - 1;
        if (ww < 0 || ww >= H28) continue;
        acc += xp[hh * H28 + ww] * w[kh * 3 + kw];
      }
    }
    op[p] = acc;
  }
}

// ---------------------------------------------------------------------------
// 4) Per-(b,c) row covariance: x2cov[h,g] = sum_w xc[h,w]*xc[g,w] / 27
// ---------------------------------------------------------------------------
__global__ void cov_kernel(const float* __restrict__ x1, float* __restrict__ x2cov) {
  __shared__ float tile[HW28];
  __shared__ float rmean[H28];
  int bc = blockIdx.x;
  const float* ip = x1 + (size_t)bc * HW28;
  float* op = x2cov + (size_t)bc * HW28;
  for (int i = threadIdx.x; i < HW28; i += 256) tile[i] = ip[i];
  __syncthreads();
  if (threadIdx.x < H28) {
    float s = 0.f;
    for (int wq = 0; wq < H28; ++wq) s += tile[threadIdx.x * H28 + wq];
    rmean[threadIdx.x] = s * (1.f / (float)H28);
  }
  __syncthreads();
  for (int i = threadIdx.x; i < HW28; i += 256) tile[i] -= rmean[i / H28];
  __syncthreads();
  for (int o = threadIdx.x; o < HW28; o += 256) {
    int h = o / H28, gg = o % H28;
    const float* rh = tile + h * H28;
    const float* rg = tile + gg * H28;
    float acc = 0.f;
#pragma unroll
    for (int wq = 0; wq < H28; ++wq) acc += rh[wq] * rg[wq];
    op[o] = acc * (1.f / (float)(H28 - 1));
  }
}

// ---------------------------------------------------------------------------
// 5) Fold SE scale into conv1_w's first column block: w'[b,o,c] = W1[o,c]*s[b,c]
// ---------------------------------------------------------------------------
__global__ void wprime_kernel(const float* __restrict__ conv1_w,  // (512,2048)
                              const float* __restrict__ s_scale,  // (b,512)
                              float* __restrict__ wprime) {       // (b,512,512)
  int t = blockIdx.x * 256 + threadIdx.x;
  if (t >= BATCH * C2 * C2) return;
  int c = t % C2;
  int o = (t / C2) % C2;
  int b = t / (C2 * C2);
  wprime[t] = conv1_w[(size_t)o * (4 * C2) + c] * s_scale[b * C2 + c];
}

// ---------------------------------------------------------------------------
// WMMA f32 GEMM with async double-buffered LDS staging.
// Block: 256 thr = 8 waves; block tile 256(M) x 64(N); wave tile 32x64
// (2x4 accumulators). K staged 16-wide; ASYNCcnt-tracked global->LDS copies.
// A tile staged row-major (b128 asyncs), B tile staged TRANSPOSED ([col][k],
// b32 asyncs) so each lane's WMMA k-pair is LDS-contiguous (ds_load_2addr_b64,
// no register repacking). Every wave issues exactly 8 asyncs per K-slab
// (OOB columns are address-clamped, never skipped), keeping ASYNCcnt exact.
// Requires: M % 256 == 0, K % 16 == 0, lda % 4 == 0, K >= 16.
// epi: 0 = raw (optionally += C), 1 = sigmoid(bn(v + bias)),
//      2 = relu(bn(v + bias + bilinear_up2(z28)))
// ---------------------------------------------------------------------------
#define LDA_S 20   // 16 + 4 pad (16B-aligned rows, conflict-free reads)
#define LDB_S 20   // transposed B: 64 cols x 16 k, stride 20

__global__ __launch_bounds__(256)
void gemm_wmma_kernel(const float* __restrict__ A, int lda, long sA,
                      const float* __restrict__ Bm, int ldb, long sB,
                      float* __restrict__ Cm, int ldc, long sC,
                      int M, int N, int K,
                      int epi, int accFlag,
                      const float* __restrict__ bias,
                      const float* __restrict__ bnG, const float* __restrict__ bnB,
                      const float* __restrict__ bnM, const float* __restrict__ bnV,
                      const float* __restrict__ z28, long sZ) {
  __shared__ float ldsA[2][256 * LDA_S];
  __shared__ float ldsB[2][64 * LDB_S];

  int bz = blockIdx.z;
  const float* Ab = A + (size_t)bz * sA;
  const float* Bb = Bm + (size_t)bz * sB;
  float* Cb = Cm + (size_t)bz * sC;
  const float* Zb = z28 ? (z28 + (size_t)bz * sZ) : nullptr;

  int tid = threadIdx.x;
  int wid = tid >> 5;    // 8 waves stacked along M
  int lane = tid & 31;
  int half = lane >> 4;
  int l16 = lane & 15;
  int mTile = blockIdx.y * 256;
  int nTile = blockIdx.x * 64;

  // staging coordinates (per thread, constant across K steps)
  int aRow[4], aC4[4];
  int bK[4], bColL[4], bColG[4];
#pragma unroll
  for (int it = 0; it < 4; ++it) {
    int idx = tid + it * 256;           // A: 0..1023 -> 256 rows x 4 chunks
    aRow[it] = idx >> 2;
    aC4[it] = (idx & 3) << 2;
    // B: 0..1023 -> 16 k-rows x 64 cols, consecutive tid = consecutive col
    bK[it] = idx >> 6;
    bColL[it] = idx & 63;
    int gc = nTile + bColL[it];
    bColG[it] = gc < N ? gc : N - 1;    // clamp: always in-bounds, never skip
  }

  v8f zero = {0.f, 0.f, 0.f, 0.f, 0.f, 0.f, 0.f, 0.f};
  v8f acc[8];
#pragma unroll
  for (int i = 0; i < 8; ++i) acc[i] = zero;

  int kSteps = K >> 4;

  // ---- prologue: stage tile 0 into buffer 0 ----
#pragma unroll
  for (int it = 0; it < 4; ++it) {
    async_copy_b128(lds_off32(&ldsA[0][aRow[it] * LDA_S + aC4[it]]),
                    Ab + (size_t)(mTile + aRow[it]) * lda + aC4[it]);
    async_copy_b32(lds_off32(&ldsB[0][bColL[it] * LDB_S + bK[it]]),
                   Bb + (size_t)bK[it] * ldb + bColG[it]);
  }

  for (int ks = 0; ks < kSteps; ++ks) {
    int cur = ks & 1;
    // ---- stage next tile into the other buffer (safe: barrier at loop end
    //      of the previous iteration guarantees no one still reads it) ----
    if (ks + 1 < kSteps) {
      int k0n = (ks + 1) << 4;
#pragma unroll
      for (int it = 0; it < 4; ++it) {
        async_copy_b128(lds_off32(&ldsA[cur ^ 1][aRow[it] * LDA_S + aC4[it]]),
                        Ab + (size_t)(mTile + aRow[it]) * lda + k0n + aC4[it]);
        async_copy_b32(lds_off32(&ldsB[cur ^ 1][bColL[it] * LDB_S + bK[it]]),
                       Bb + (size_t)(k0n + bK[it]) * ldb + bColG[it]);
      }
      // 8 newest (next tile) may remain outstanding; current tile is done.
      asm volatile("s_wait_asynccnt 0x8" ::: "memory");
    } else {
      asm volatile("s_wait_asynccnt 0x0" ::: "memory");
    }
    __syncthreads();

    const float* lA = &ldsA[cur][0];
    const float* lB = &ldsB[cur][0];
#pragma unroll
    for (int kk = 0; kk < 16; kk += 4) {
      v2f af[2], bf[4];
#pragma unroll
      for (int mi = 0; mi < 2; ++mi) {
        int base = (wid * 32 + mi * 16 + l16) * LDA_S + kk + 2 * half;
        af[mi].x = lA[base];
        af[mi].y = lA[base + 1];
      }
#pragma unroll
      for (int ni = 0; ni < 4; ++ni) {
        int base = (ni * 16 + l16) * LDB_S + kk + 2 * half;
        bf[ni].x = lB[base];
        bf[ni].y = lB[base + 1];
      }
#pragma unroll
      for (int mi = 0; mi < 2; ++mi)
#pragma unroll
        for (int ni = 0; ni < 4; ++ni)
          acc[mi * 4 + ni] = wmma4(af[mi], bf[ni], acc[mi * 4 + ni]);
    }
    __syncthreads();
  }

  // --- epilogue ---
#pragma unroll
  for (int mi = 0; mi < 2; ++mi) {
#pragma unroll
    for (int ni = 0; ni < 4; ++ni) {
      v8f a = acc[mi * 4 + ni];
      int col = nTile + ni * 16 + l16;
      if (col >= N) continue;
      int rowBase = mTile + wid * 32 + mi * 16 + half * 8;
#pragma unroll
      for (int i = 0; i < 8; ++i) {
        int row = rowBase + i;
        float v = a[i];
        size_t oi = (size_t)row * ldc + col;
        if (epi == 0) {
          if (accFlag) v += Cb[oi];
          Cb[oi] = v;
        } else if (epi == 1) {
          v += bias[row];
          v = (v - bnM[row]) * rsqrtf(bnV[row] + BN_EPS) * bnG[row] + bnB[row];
          Cb[oi] = 1.f / (1.f + expf(-v));
        } else {
          v += bias[row];
          // bilinear 2x upsample of z28 (half-pixel, edge clamp)
          int hh = col / H56, ww = col % H56;
          float fy = 0.5f * hh - 0.25f; if (fy < 0.f) fy = 0.f;
          float fx = 0.5f * ww - 0.25f; if (fx < 0.f) fx = 0.f;
          int y0 = (int)fy, x0 = (int)fx;
          int y1 = (y0 + 1 < H28) ? y0 + 1 : H28 - 1;
          int x1 = (x0 + 1 < H28) ? x0 + 1 : H28 - 1;
          float ty = fy - (float)y0, tx = fx - (float)x0;
          const float* zp = Zb + (size_t)row * HW28;
          float z00 = zp[y0 * H28 + x0], z01 = zp[y0 * H28 + x1];
          float z10 = zp[y1 * H28 + x0], z11 = zp[y1 * H28 + x1];
          v += (z00 * (1.f - tx) + z01 * tx) * (1.f - ty) +
               (z10 * (1.f - tx) + z11 * tx) * ty;
          v = (v - bnM[row]) * rsqrtf(bnV[row] + BN_EPS) * bnG[row] + bnB[row];
          Cb[oi] = v > 0.f ? v : 0.f;
        }
      }
    }
  }
}

// ---------------------------------------------------------------------------
extern "C" void kernel_launch(void* const* d_in, const int* in_sizes, int n_in,
                              void* d_out, int out_size, void* d_ws, size_t ws_size,
                              hipStream_t stream) {
  (void)in_sizes; (void)n_in; (void)out_size; (void)ws_size;
  const float* y       = (const float*)d_in[0];
  const float* x       = (const float*)d_in[1];
  const float* fc1_w   = (const float*)d_in[2];
  const float* fc2_w   = (const float*)d_in[3];
  const float* conv_w  = (const float*)d_in[4];
  const float* conv_b  = (const float*)d_in[5];
  const float* conv1_w = (const float*)d_in[6];
  const float* conv1_b = (const float*)d_in[7];
  const float* bn_g    = (const float*)d_in[8];
  const float* bn_b    = (const float*)d_in[9];
  const float* bn_m    = (const float*)d_in[10];
  const float* bn_v    = (const float*)d_in[11];
  const float* dw_w    = (const float*)d_in[12];
  const float* dw_b    = (const float*)d_in[13];
  float* out = (float*)d_out;

  float* ws = (float*)d_ws;
  size_t off = 0;
  float* s_gap   = ws + off; off += (size_t)BATCH * C2;
  float* s_scale = ws + off; off += (size_t)BATCH * C2;
  float* x1      = ws + off; off += (size_t)BATCH * C_IN * HW28;
  float* x2cov   = ws + off; off += (size_t)BATCH * C_IN * HW28;
  float* x2s     = ws + off; off += (size_t)BATCH * C2 * HW28;
  float* z28     = ws + off; off += (size_t)BATCH * C2 * HW28;
  float* wprime  = ws + off; off += (size_t)BATCH * C2 * C2;
  // total ~21.5M floats (~86 MB) of workspace

  gap_kernel<<<BATCH * C2, 256, 0, stream>>>(y, s_gap);
  se_fc_kernel<<<BATCH, 256, 0, stream>>>(s_gap, fc1_w, fc2_w, s_scale);
  dwconv_kernel<<<BATCH * C_IN, 256, 0, stream>>>(x, dw_w, dw_b, x1);
  cov_kernel<<<BATCH * C_IN, 256, 0, stream>>>(x1, x2cov);
  wprime_kernel<<<(BATCH * C2 * C2 + 255) / 256, 256, 0, stream>>>(conv1_w, s_scale, wprime);

  // G2: x2s = sigmoid(bn(conv_w @ x2cov + conv_b))     (512x1024)@(1024x784)
  {
    dim3 grid((HW28 + 63) / 64, C2 / 256, BATCH);
    gemm_wmma_kernel<<<grid, 256, 0, stream>>>(
        conv_w, C_IN, 0L, x2cov, HW28, (long)C_IN * HW28,
        x2s, HW28, (long)C2 * HW28, C2, HW28, C_IN,
        1, 0, conv_b, bn_g, bn_b, bn_m, bn_v, nullptr, 0L);
  }
  // G3a: z28 = conv1_w[:,512:1024] @ x2s               (512x512)@(512x784)
  {
    dim3 grid((HW28 + 63) / 64, C2 / 256, BATCH);
    gemm_wmma_kernel<<<grid, 256, 0, stream>>>(
        conv1_w + C2, 4 * C2, 0L, x2s, HW28, (long)C2 * HW28,
        z28, HW28, (long)C2 * HW28, C2, HW28, C2,
        0, 0, nullptr, nullptr, nullptr, nullptr, nullptr, nullptr, 0L);
  }
  // G3b: z28 += conv1_w[:,1024:2048] @ x               (512x1024)@(1024x784)
  {
    dim3 grid((HW28 + 63) / 64, C2 / 256, BATCH);
    gemm_wmma_kernel<<<grid, 256, 0, stream>>>(
        conv1_w + 2 * C2, 4 * C2, 0L, x, HW28, (long)C_IN * HW28,
        z28, HW28, (long)C2 * HW28, C2, HW28, C_IN,
        0, 1, nullptr, nullptr, nullptr, nullptr, nullptr, nullptr, 0L);
  }
  // G4: out = relu(bn(wprime @ y + conv1_b + up2(z28)))  (512x512)@(512x3136)
  {
    dim3 grid(HW56 / 64, C2 / 256, BATCH);
    gemm_wmma_kernel<<<grid, 256, 0, stream>>>(
        wprime, C2, (long)C2 * C2, y, HW56, (long)C2 * HW56,
        out, HW56, (long)C2 * HW56, C2, HW56, C2,
        2, 0, conv1_b, bn_g, bn_b, bn_m, bn_v, z28, (long)C2 * HW28);
  }
}